// LSTM_8263517078098
// MI455X (gfx1250) — compile-verified
//
#include <hip/hip_runtime.h>
#include <math.h>

// ---------------------------------------------------------------------------
// Problem constants (fixed by the reference)
// ---------------------------------------------------------------------------
#define T_STEPS 4096
#define DIN     64
#define H       64
#define G4H     256     // 4*H
#define NLAYERS 3
#define LIN     1024
#define NOUT    8
#define FLAT    (T_STEPS * H)       // 262144
#define ROWS_OUT (NOUT * T_STEPS)   // 32768
#define WPITCH  68      // padded LDS row pitch (floats): bank-start 4r mod 64

typedef __attribute__((ext_vector_type(2))) float v2f;
typedef __attribute__((ext_vector_type(8))) float v8f;

// ---------------------------------------------------------------------------
// Kernel 1: xw0 = x(T x 64) @ Wih0(256 x 64)^T + (bih0 + bhh0)   -> (T x 256)
// One wave per 16x16 tile; K=64 swept by 16 V_WMMA_F32_16X16X4_F32.
// ---------------------------------------------------------------------------
__global__ __launch_bounds__(256) void xw_gemm_wmma(
    const float* __restrict__ in,    // (T, 64)
    const float* __restrict__ Wih,   // (256, 64) layer 0
    const float* __restrict__ bih,   // (256,)
    const float* __restrict__ bhh,   // (256,)
    float* __restrict__ XW)          // (T, 256)
{
    const int lane = threadIdx.x & 31;
    const int wave = threadIdx.x >> 5;
    const int tile = blockIdx.x * 8 + wave;        // 256*16 = 4096 tiles
    const int m0 = (tile >> 4) << 4;
    const int n0 = (tile & 15) << 4;
    const int r  = lane & 15;
    const int hi = lane >> 4;                      // 0: K=0,1  1: K=2,3

    const float* ap = in  + (m0 + r) * DIN + 2 * hi;   // A rows from x
    const float* bp = Wih + (n0 + r) * DIN + 2 * hi;   // B[k][n] = Wih[n][k]

    v8f acc = {};
#pragma unroll
    for (int k0 = 0; k0 < DIN; k0 += 4) {
        v2f a = *(const v2f*)(ap + k0);
        v2f b = *(const v2f*)(bp + k0);
        acc = __builtin_amdgcn_wmma_f32_16x16x4_f32(
            false, a, false, b, (short)0, acc, false, false);
    }

    const int n    = n0 + r;
    const float bs = bih[n] + bhh[n];
#pragma unroll
    for (int v = 0; v < 8; ++v) {
        XW[(size_t)(m0 + v + 8 * hi) * G4H + n] = acc[v] + bs;
    }
}

// ---------------------------------------------------------------------------
// Kernel 2: fully fused 3-layer LSTM with wavefront pipelining.
// 768 threads = 3 layer-groups x 256. Pipeline step s computes
//   L0: h0[s], L1: h1[s-1], L2: h2[s-2]  (serial chain 4098 steps, not 12288).
// Whh row per thread in 64 VGPRs; Wih (layers 1,2) LDS-resident with padded
// pitch; h double-buffered in LDS (producers write `cur`, consumers read
// `prev`); 4 independent accumulators per dot to cut the fp32 FMA chain.
// ---------------------------------------------------------------------------
__device__ __forceinline__ float sigm(float x) { return 1.0f / (1.0f + __expf(-x)); }

__global__ __launch_bounds__(768) void lstm_fused(
    const float* __restrict__ XW,    // (T, 256) layer-0 pre-activations (+bias)
    const float* __restrict__ Wih,   // (3, 256, 64)
    const float* __restrict__ Whh,   // (3, 256, 64)
    const float* __restrict__ bih,   // (3, 256)
    const float* __restrict__ bhh,   // (3, 256)
    float* __restrict__ h2out)       // (T, 64) final-layer hidden sequence
{
    __shared__ __align__(16) float wih_s[2][G4H * WPITCH];   // layers 1,2
    __shared__ __align__(16) float h_s[NLAYERS][2][H];       // double-buffered
    __shared__ float g_s[NLAYERS][G4H];

    const int tid   = threadIdx.x;
    const int layer = tid >> 8;        // uniform per wave (8 waves per layer)
    const int r     = tid & 255;

    // Stage Wih for layers 1,2 into LDS (padded rows, b128 copies)
    for (int idx = tid; idx < 2 * G4H; idx += 768) {
        const int li = idx >> 8, row = idx & 255;
        const float4* src = (const float4*)(Wih + (size_t)(li + 1) * G4H * H
                                                + (size_t)row * H);
        float4* dst = (float4*)&wih_s[li][row * WPITCH];
#pragma unroll
        for (int q = 0; q < H / 4; ++q) dst[q] = src[q];
    }
    if (tid < NLAYERS * 2 * H) ((float*)h_s)[tid] = 0.0f;

    // Per-thread recurrent weight row in VGPRs
    float w[H];
#pragma unroll
    for (int k = 0; k < H; ++k)
        w[k] = Whh[(size_t)layer * G4H * H + (size_t)r * H + k];

    const float bsum = (layer == 0) ? 0.0f
                      : (bih[layer * G4H + r] + bhh[layer * G4H + r]);
    float c = 0.0f;
    float xw_cur = (layer == 0) ? XW[r] : 0.0f;   // layer-0 row for t = 0
    __syncthreads();

    for (int s = 0; s < T_STEPS + NLAYERS - 1; ++s) {
        const int  t    = s - layer;
        const bool act  = (t >= 0) && (t < T_STEPS);
        const int  prev = (s + 1) & 1;
        const int  cur  = s & 1;

        if (act) {
            float a0 = 0, a1 = 0, a2 = 0, a3 = 0;
            const float* hp = &h_s[layer][prev][0];      // LDS broadcast reads
#pragma unroll
            for (int k = 0; k < H; k += 4) {
                const float4 hv = *(const float4*)&hp[k];
                a0 += w[k] * hv.x; a1 += w[k + 1] * hv.y;
                a2 += w[k + 2] * hv.z; a3 += w[k + 3] * hv.w;
            }
            float g;
            if (layer == 0) {
                g = xw_cur + ((a0 + a1) + (a2 + a3));
            } else {
                float b0 = 0, b1 = 0, b2 = 0, b3 = 0;
                const float* hi = &h_s[layer - 1][prev][0];
                const float* wr = &wih_s[layer - 1][r * WPITCH];
#pragma unroll
                for (int k = 0; k < H; k += 4) {
                    const float4 wv = *(const float4*)&wr[k];
                    const float4 hv = *(const float4*)&hi[k];
                    b0 += wv.x * hv.x; b1 += wv.y * hv.y;
                    b2 += wv.z * hv.z; b3 += wv.w * hv.w;
                }
                g = bsum + ((a0 + a1) + (a2 + a3)) + ((b0 + b1) + (b2 + b3));
            }
            g_s[layer][r] = g;
        }
        // Software-prefetch next layer-0 xw row across barrier+gate latency
        float xw_nxt = 0.0f;
        if (layer == 0 && (s + 1) < T_STEPS)
            xw_nxt = XW[(size_t)(s + 1) * G4H + r];
        __syncthreads();

        if (act && r < H) {
            const float gi = g_s[layer][r];
            const float gf = g_s[layer][H + r];
            const float gg = g_s[layer][2 * H + r];
            const float go = g_s[layer][3 * H + r];
            c = sigm(gf) * c + sigm(gi) * tanhf(gg);
            const float h = sigm(go) * tanhf(c);
            h_s[layer][cur][r] = h;
            if (layer == NLAYERS - 1) h2out[(size_t)t * H + r] = h;
        }
        __syncthreads();
        xw_cur = xw_nxt;
    }
}

// ---------------------------------------------------------------------------
// Kernel 3: block-per-row GEMV (W1: 1024 x 262144, 1 GB stream -> HBM roofline)
// ---------------------------------------------------------------------------
__global__ __launch_bounds__(256) void gemv_block(
    const float* __restrict__ W, const float* __restrict__ x,
    const float* __restrict__ b, float* __restrict__ y, int cols4)
{
    const int row = blockIdx.x;
    const int tid = threadIdx.x;
    const float4* __restrict__ Wr = (const float4*)W + (size_t)row * cols4;
    const float4* __restrict__ xv = (const float4*)x;

    float4 acc4 = {0.f, 0.f, 0.f, 0.f};
    for (int i = tid; i < cols4; i += 256) {
        __builtin_prefetch(Wr + i + 4096, 0, 0);   // global_prefetch
        const float4 wv = Wr[i];
        const float4 vv = xv[i];
        acc4.x += wv.x * vv.x; acc4.y += wv.y * vv.y;
        acc4.z += wv.z * vv.z; acc4.w += wv.w * vv.w;
    }
    float acc = (acc4.x + acc4.y) + (acc4.z + acc4.w);
#pragma unroll
    for (int off = 16; off > 0; off >>= 1) acc += __shfl_xor(acc, off, 32);

    __shared__ float red[8];
    if ((tid & 31) == 0) red[tid >> 5] = acc;
    __syncthreads();
    if (tid == 0) {
        float s = 0.0f;
#pragma unroll
        for (int i = 0; i < 8; ++i) s += red[i];
        y[row] = s + b[row];
    }
}

// ---------------------------------------------------------------------------
// Kernel 4: wave-per-row GEMV (W2 1024x1024, Wout 32768x1024)
// ---------------------------------------------------------------------------
__global__ __launch_bounds__(256) void gemv_wave(
    const float* __restrict__ W, const float* __restrict__ x,
    const float* __restrict__ b, float* __restrict__ y,
    int rows, int cols4)
{
    const int lane = threadIdx.x & 31;
    const int gw = (int)((blockIdx.x * blockDim.x + threadIdx.x) >> 5);
    if (gw >= rows) return;

    const float4* __restrict__ Wr = (const float4*)W + (size_t)gw * cols4;
    const float4* __restrict__ xv = (const float4*)x;

    float4 acc4 = {0.f, 0.f, 0.f, 0.f};
    for (int i = lane; i < cols4; i += 32) {
        const float4 wv = Wr[i];
        const float4 vv = xv[i];
        acc4.x += wv.x * vv.x; acc4.y += wv.y * vv.y;
        acc4.z += wv.z * vv.z; acc4.w += wv.w * vv.w;
    }
    float acc = (acc4.x + acc4.y) + (acc4.z + acc4.w);
#pragma unroll
    for (int off = 16; off > 0; off >>= 1) acc += __shfl_xor(acc, off, 32);
    if (lane == 0) y[gw] = acc + b[gw];
}

// ---------------------------------------------------------------------------
// Host-side launch sequence (all on `stream`, graph-capture safe)
// ---------------------------------------------------------------------------
extern "C" void kernel_launch(void* const* d_in, const int* in_sizes, int n_in,
                              void* d_out, int out_size, void* d_ws, size_t ws_size,
                              hipStream_t stream)
{
    const float* x    = (const float*)d_in[0];   // (T, DIN)
    const float* Wih  = (const float*)d_in[1];   // (L, 256, 64)
    const float* Whh  = (const float*)d_in[2];   // (L, 256, 64)
    const float* bih  = (const float*)d_in[3];   // (L, 256)
    const float* bhh  = (const float*)d_in[4];   // (L, 256)
    const float* W1   = (const float*)d_in[5];   // (1024, 262144)
    const float* b1   = (const float*)d_in[6];
    const float* W2   = (const float*)d_in[7];   // (1024, 1024)
    const float* b2   = (const float*)d_in[8];
    const float* Wout = (const float*)d_in[9];   // (32768, 1024)
    const float* bout = (const float*)d_in[10];
    float* out = (float*)d_out;                  // (T, NOUT) flat

    // Workspace layout (floats): xw(4MB) | h2(1MB) | y1 | y2
    float* xw = (float*)d_ws;
    float* h2 = xw + (size_t)T_STEPS * G4H;
    float* y1 = h2 + (size_t)T_STEPS * H;
    float* y2 = y1 + LIN;

    // Layer-0 input projection: WMMA GEMM (the only non-sequential LSTM work
    // against external input; layers 1,2 consume h in-kernel).
    xw_gemm_wmma<<<(T_STEPS / 16) * (G4H / 16) / 8, 256, 0, stream>>>(
        x, Wih, bih, bhh, xw);

    // Fused, pipelined 3-layer recurrence: serial chain 4098 steps.
    lstm_fused<<<1, 768, 0, stream>>>(xw, Wih, Whh, bih, bhh, h2);

    // y1 = W1 @ flat + b1   (1 GB weight stream -> HBM roofline)
    gemv_block<<<LIN, 256, 0, stream>>>(W1, h2, b1, y1, FLAT / 4);
    // y2 = W2 @ y1 + b2
    gemv_wave<<<LIN / 8, 256, 0, stream>>>(W2, y1, b2, y2, LIN, LIN / 4);
    // out = Wout @ y2 + bout (flat order == (T, NOUT) row-major reshape)
    gemv_wave<<<ROWS_OUT / 8, 256, 0, stream>>>(Wout, y2, bout, out,
                                                ROWS_OUT, LIN / 4);
}